// GAT_81570018885998
// MI455X (gfx1250) — compile-verified
//
#include <hip/hip_runtime.h>
#include <math.h>

#define N_NODES 40000
#define N_EDGES 640000
#define NE_TOT  (N_EDGES + N_NODES)   /* 680000 edges incl. self loops */
#define F_IN    128
#define HID     64
#define HEADS   4
#define C1      (HEADS*HID)           /* 256 */
#define NCLS    40
#define NCLSP   48                    /* padded cols for layer-2 GEMM */
#define SLOPE   0.2f

typedef float v2f __attribute__((ext_vector_type(2)));
typedef float v8f __attribute__((ext_vector_type(8)));

// ---------------- helpers ----------------

__device__ __forceinline__ float lrelu(float x) { return x > 0.0f ? x : SLOPE * x; }

// order-invariant (deterministic) float atomic max via integer atomics
__device__ __forceinline__ void atomic_max_f32(float* addr, float val) {
  if (val >= 0.0f) {
    atomicMax((int*)addr, __float_as_int(val));
  } else {
    atomicMin((unsigned int*)addr, __float_as_uint(val));
  }
}

__device__ __forceinline__ void edge_sd(const int* __restrict__ ei, int e, int& s, int& d) {
  if (e < N_EDGES) { s = ei[e]; d = ei[N_EDGES + e]; }
  else             { s = e - N_EDGES; d = s; }       // appended self loops
}

// ---------------- init / pad ----------------

__global__ void k_fill(float* __restrict__ p, int n, float v) {
  int i = blockIdx.x * blockDim.x + threadIdx.x;
  if (i < n) p[i] = v;
}

// W2 [256 x 40] -> W2p [256 x 48], zero padded (removes guards from GEMM2 loop)
__global__ void k_pad_w2(const float* __restrict__ W2, float* __restrict__ W2p) {
  int t = blockIdx.x * blockDim.x + threadIdx.x;
  if (t >= C1 * NCLSP) return;
  int r = t / NCLSP, c = t % NCLSP;
  W2p[t] = (c < NCLS) ? W2[r * NCLS + c] : 0.0f;
}

// ---------------- fp32 WMMA GEMM: C[MxN] = A[MxK] @ B[KxN] ----------------
// One wave per 16x16 C tile; V_WMMA_F32_16X16X4_F32, K fully unrolled in steps of 4.
// A-frag (ISA 7.12.2): lane L<16: M=L, V0=K0,V1=K1 ; lane L+16: M=L, V0=K2,V1=K3.
// B-frag mirrors with N across lanes. C/D: VGPR r -> M=r (lanes 0-15), M=8+r (16-31).
// No guards inside the loop: B must be padded so all 16 cols are readable.

template <int K, int LDB, int LDC, int NV>
__global__ void k_gemm_wmma(const float* __restrict__ A, const float* __restrict__ B,
                            float* __restrict__ C) {
  const int lane = threadIdx.x;      // blockDim.x == 32
  const int mn   = lane & 15;
  const int half = lane >> 4;        // 0 or 1
  const int row0 = blockIdx.y * 16;
  const int col0 = blockIdx.x * 16;

  v8f acc = {};
  const float* ap = A + (size_t)(row0 + mn) * K + 2 * half;      // += k0
  const float* bp = B + (size_t)(2 * half) * LDB + col0 + mn;    // += k0*LDB

#pragma unroll
  for (int k0 = 0; k0 < K; k0 += 4) {
    v2f a = *(const v2f*)(ap + k0);                 // 8B-aligned (even offset)
    v2f b;
    b.x = bp[(size_t)k0 * LDB];
    b.y = bp[(size_t)(k0 + 1) * LDB];
    acc = __builtin_amdgcn_wmma_f32_16x16x4_f32(
        /*neg_a=*/false, a, /*neg_b=*/false, b,
        /*c_mod=*/(short)0, acc, /*reuse_a=*/false, /*reuse_b=*/false);
  }

#pragma unroll
  for (int r = 0; r < 8; ++r) {
    const int row = row0 + r + 8 * half;
    const int col = col0 + mn;
    if (NV == LDC || col < NV) C[(size_t)row * LDC + col] = acc[r];
  }
}

// ---------------- attention scalars ----------------

__global__ void k_att1(const float* __restrict__ H, const float* __restrict__ att_s,
                       const float* __restrict__ att_d, float* __restrict__ as,
                       float* __restrict__ ad) {
  int t = blockIdx.x * blockDim.x + threadIdx.x;
  if (t >= N_NODES * HEADS) return;
  int n = t >> 2, h = t & 3;
  const float* hp = H + (size_t)n * C1 + h * HID;
  const float* sp = att_s + h * HID;
  const float* dp = att_d + h * HID;
  float s = 0.f, d = 0.f;
  for (int c = 0; c < HID; ++c) { float v = hp[c]; s += v * sp[c]; d += v * dp[c]; }
  as[t] = s; ad[t] = d;
}

__global__ void k_att2(const float* __restrict__ H, const float* __restrict__ att_s,
                       const float* __restrict__ att_d, float* __restrict__ as,
                       float* __restrict__ ad) {
  int n = blockIdx.x * blockDim.x + threadIdx.x;
  if (n >= N_NODES) return;
  const float* hp = H + (size_t)n * NCLS;
  float s = 0.f, d = 0.f;
  for (int c = 0; c < NCLS; ++c) { float v = hp[c]; s += v * att_s[c]; d += v * att_d[c]; }
  as[n] = s; ad[n] = d;
}

// ---------------- edge passes (segment softmax) ----------------

template <int H>
__global__ void k_edge_max(const int* __restrict__ ei, const float* __restrict__ as,
                           const float* __restrict__ ad, float* __restrict__ m) {
  int t = blockIdx.x * blockDim.x + threadIdx.x;
  if (t >= NE_TOT * H) return;
  int e = t / H, h = t % H;
  int s, d; edge_sd(ei, e, s, d);
  float v = lrelu(as[s * H + h] + ad[d * H + h]);
  atomic_max_f32(&m[d * H + h], v);
}

template <int H>
__global__ void k_edge_sum(const int* __restrict__ ei, const float* __restrict__ as,
                           const float* __restrict__ ad, const float* __restrict__ m,
                           float* __restrict__ z) {
  int t = blockIdx.x * blockDim.x + threadIdx.x;
  if (t >= NE_TOT * H) return;
  int e = t / H, h = t % H;
  int s, d; edge_sd(ei, e, s, d);
  float v = lrelu(as[s * H + h] + ad[d * H + h]);
  atomicAdd(&z[d * H + h], expf(v - m[d * H + h]));
}

// layer-1 aggregation: one wave per (edge, head); 64 channels -> 2 per lane
__global__ void k_edge_agg1(const int* __restrict__ ei, const float* __restrict__ as,
                            const float* __restrict__ ad, const float* __restrict__ m,
                            const float* __restrict__ z, const float* __restrict__ H1,
                            float* __restrict__ out) {
  int g    = (blockIdx.x * blockDim.x + threadIdx.x) >> 5;
  int lane = threadIdx.x & 31;
  if (g >= NE_TOT * HEADS) return;
  int e = g >> 2, h = g & 3;
  int s, d; edge_sd(ei, e, s, d);
  float v     = lrelu(as[s * HEADS + h] + ad[d * HEADS + h]);
  float alpha = expf(v - m[d * HEADS + h]) / z[d * HEADS + h];
  const float* hs = H1 + (size_t)s * C1 + h * HID;
  float*       op = out + (size_t)d * C1 + h * HID;
  atomicAdd(&op[lane],      hs[lane]      * alpha);
  atomicAdd(&op[lane + 32], hs[lane + 32] * alpha);
}

// layer-2 aggregation: one wave per edge; 40 channels
__global__ void k_edge_agg2(const int* __restrict__ ei, const float* __restrict__ as,
                            const float* __restrict__ ad, const float* __restrict__ m,
                            const float* __restrict__ z, const float* __restrict__ H2,
                            float* __restrict__ out) {
  int e    = (blockIdx.x * blockDim.x + threadIdx.x) >> 5;
  int lane = threadIdx.x & 31;
  if (e >= NE_TOT) return;
  int s, d; edge_sd(ei, e, s, d);
  float v     = lrelu(as[s] + ad[d]);
  float alpha = expf(v - m[d]) / z[d];
  const float* hs = H2 + (size_t)s * NCLS;
  float*       op = out + (size_t)d * NCLS;
  atomicAdd(&op[lane], hs[lane] * alpha);
  if (lane < NCLS - 32) atomicAdd(&op[lane + 32], hs[lane + 32] * alpha);
}

// ---------------- elementwise ----------------

__global__ void k_bias_relu(float* __restrict__ o, const float* __restrict__ b) {
  int i = blockIdx.x * blockDim.x + threadIdx.x;
  if (i >= N_NODES * C1) return;
  float v = o[i] + b[i & (C1 - 1)];
  o[i] = v > 0.0f ? v : 0.0f;
}

// add b2 + in-place log_softmax over 40 classes; one wave per node
__global__ void k_lsm(float* __restrict__ out, const float* __restrict__ b2) {
  int wid  = threadIdx.x >> 5;
  int lane = threadIdx.x & 31;
  int n    = blockIdx.x * (blockDim.x >> 5) + wid;
  if (n >= N_NODES) return;
  float* row = out + (size_t)n * NCLS;
  float v0 = row[lane] + b2[lane];
  float v1 = (lane < NCLS - 32) ? (row[lane + 32] + b2[lane + 32]) : -INFINITY;
  float mx = fmaxf(v0, v1);
#pragma unroll
  for (int off = 16; off > 0; off >>= 1) mx = fmaxf(mx, __shfl_xor(mx, off, 32));
  float sm = expf(v0 - mx) + ((lane < NCLS - 32) ? expf(v1 - mx) : 0.0f);
#pragma unroll
  for (int off = 16; off > 0; off >>= 1) sm += __shfl_xor(sm, off, 32);
  float ls = mx + logf(sm);
  row[lane] = v0 - ls;
  if (lane < NCLS - 32) row[lane + 32] = v1 - ls;
}

// ---------------- launch ----------------

extern "C" void kernel_launch(void* const* d_in, const int* in_sizes, int n_in,
                              void* d_out, int out_size, void* d_ws, size_t ws_size,
                              hipStream_t stream) {
  const float* x    = (const float*)d_in[0];
  const int*   ei   = (const int*)d_in[1];
  const float* W1   = (const float*)d_in[2];
  const float* as1w = (const float*)d_in[3];
  const float* ad1w = (const float*)d_in[4];
  const float* b1   = (const float*)d_in[5];
  const float* W2   = (const float*)d_in[6];
  const float* as2w = (const float*)d_in[7];
  const float* ad2w = (const float*)d_in[8];
  const float* b2   = (const float*)d_in[9];
  float* out = (float*)d_out;

  float* ws = (float*)d_ws;
  size_t off = 0;
  float* H1  = ws + off; off += (size_t)N_NODES * C1;    // x @ W1
  float* O1  = ws + off; off += (size_t)N_NODES * C1;    // layer-1 aggregated / relu'd
  float* H2  = ws + off; off += (size_t)N_NODES * NCLS;  // O1 @ W2
  float* W2p = ws + off; off += (size_t)C1 * NCLSP;      // zero-padded W2
  float* as1 = ws + off; off += N_NODES * HEADS;
  float* ad1 = ws + off; off += N_NODES * HEADS;
  float* m1  = ws + off; off += N_NODES * HEADS;
  float* z1  = ws + off; off += N_NODES * HEADS;
  float* as2 = ws + off; off += N_NODES;
  float* ad2 = ws + off; off += N_NODES;
  float* m2  = ws + off; off += N_NODES;
  float* z2  = ws + off; off += N_NODES;
  (void)in_sizes; (void)n_in; (void)out_size; (void)ws_size;

  const int TB = 256;
  auto blks = [](long long n, int tb) { return (unsigned)((n + tb - 1) / tb); };

  // --- init accumulators (d_out/ws are stale between replays) ---
  k_fill<<<blks((long long)N_NODES * C1, TB), TB, 0, stream>>>(O1, N_NODES * C1, 0.0f);
  k_fill<<<blks(N_NODES * HEADS, TB), TB, 0, stream>>>(m1, N_NODES * HEADS, -INFINITY);
  k_fill<<<blks(N_NODES * HEADS, TB), TB, 0, stream>>>(z1, N_NODES * HEADS, 0.0f);
  k_fill<<<blks(N_NODES, TB), TB, 0, stream>>>(m2, N_NODES, -INFINITY);
  k_fill<<<blks(N_NODES, TB), TB, 0, stream>>>(z2, N_NODES, 0.0f);
  k_fill<<<blks((long long)N_NODES * NCLS, TB), TB, 0, stream>>>(out, N_NODES * NCLS, 0.0f);
  k_pad_w2<<<blks(C1 * NCLSP, TB), TB, 0, stream>>>(W2, W2p);

  // --- layer 1 ---
  k_gemm_wmma<F_IN, C1, C1, C1>
      <<<dim3(C1 / 16, N_NODES / 16), 32, 0, stream>>>(x, W1, H1);
  k_att1<<<blks(N_NODES * HEADS, TB), TB, 0, stream>>>(H1, as1w, ad1w, as1, ad1);
  k_edge_max<HEADS><<<blks((long long)NE_TOT * HEADS, TB), TB, 0, stream>>>(ei, as1, ad1, m1);
  k_edge_sum<HEADS><<<blks((long long)NE_TOT * HEADS, TB), TB, 0, stream>>>(ei, as1, ad1, m1, z1);
  k_edge_agg1<<<blks((long long)NE_TOT * HEADS * 32, TB), TB, 0, stream>>>(ei, as1, ad1, m1, z1, H1, O1);
  k_bias_relu<<<blks((long long)N_NODES * C1, TB), TB, 0, stream>>>(O1, b1);

  // --- layer 2 ---
  k_gemm_wmma<C1, NCLSP, NCLS, NCLS>
      <<<dim3(NCLSP / 16, N_NODES / 16), 32, 0, stream>>>(O1, W2p, H2);
  k_att2<<<blks(N_NODES, TB), TB, 0, stream>>>(H2, as2w, ad2w, as2, ad2);
  k_edge_max<1><<<blks(NE_TOT, TB), TB, 0, stream>>>(ei, as2, ad2, m2);
  k_edge_sum<1><<<blks(NE_TOT, TB), TB, 0, stream>>>(ei, as2, ad2, m2, z2);
  k_edge_agg2<<<blks((long long)NE_TOT * 32, TB), TB, 0, stream>>>(ei, as2, ad2, m2, z2, H2, out);
  k_lsm<<<blks((long long)N_NODES * 32, TB), TB, 0, stream>>>(out, b2);
}